// PushingEnv_65635690218004
// MI455X (gfx1250) — compile-verified
//
#include <hip/hip_runtime.h>

// PushingEnv batched scan for MI455X (gfx1250, wave32).
//
// Roofline: streams u_seq (131072*200*4 = 105 MB) exactly once -> ~4.7 us
// floor at 23.3 TB/s HBM; VALU recurrence (~8 ops/step) is below that.
// The recurrence is nonlinear (contact clamp), so WMMA does not apply; the
// CDNA5 path that matters is async global->LDS DMA: double-buffered 256x20
// tiles of u_seq via global_load_async_to_lds_b128 + s_wait_asynccnt,
// integrated from LDS with float2 (ds_load_b64) reads.
//
// LDS layout: row stride 20 floats (80 B, 16B-aligned for b128 writes).
// tid*20 mod 64 = 4*(5*tid mod 16) -> at most 2-way read bank conflict.

#define HSTEPS   200
#define TCHUNK   20
#define NCHUNK   (HSTEPS / TCHUNK)       // 10
#define ROWS     256
#define STRIDEF  TCHUNK                   // 20 floats, no padding
#define BUFFLOATS (ROWS * STRIDEF)        // 5120 floats = 20480 B per buffer
#define GROUPS   (TCHUNK / 4)             // 5 x b128 (4-float) groups per row
#define NLOADS   GROUPS                   // 5 async loads per thread per chunk

__device__ __forceinline__ void sim_step(float u, float& x1, float& v1,
                                         float& x2, float& v2) {
    // penetration = x1 - x2 - 1 ; contact = max(pen, 0)
    // F = -K*contact - C*(v1 - v2)          (K=1e4, C=1, DT=0.01)
    // v1 += (u + F)*DT ; v2 += (-F)*DT ; x1 += v1*DT ; x2 += v2*DT
    float pen     = x1 - x2 - 1.0f;
    float contact = fmaxf(pen, 0.0f);
    float relv    = v1 - v2;
    float F       = fmaf(-10000.0f, contact, -relv);
    v1 = fmaf(u + F, 0.01f, v1);
    v2 = fmaf(F, -0.01f, v2);
    x1 = fmaf(v1, 0.01f, x1);
    x2 = fmaf(v2, 0.01f, x2);
}

__global__ __launch_bounds__(ROWS)
void pushing_env_scan_kernel(const float* __restrict__ u_seq,
                             const float* __restrict__ x1_0,
                             const float* __restrict__ v1_0,
                             const float* __restrict__ x2_0,
                             const float* __restrict__ v2_0,
                             const float* __restrict__ goal,
                             float* __restrict__ out, int B) {
    __shared__ float tile[2 * BUFFLOATS];

    const int tid = threadIdx.x;
    const int r0  = blockIdx.x * ROWS;
    const int b   = r0 + tid;

    // Low 32 bits of the flat address of a __shared__ object are the LDS
    // byte address (ISA 10.2).
    const unsigned lds_base = (unsigned)(uintptr_t)(&tile[0]);

    // Issue all async b128 loads for time-chunk c into buffer (c&1).
    // row/g math is chunk-invariant -> LICM hoists it; per chunk this is
    // just an immediate-style add per load.
    auto issue = [&](int c) {
#pragma unroll
        for (int k = 0; k < NLOADS; ++k) {
            int i   = tid + k * ROWS;             // 0..1279
            int row = i / GROUPS;                 // 0..255
            int g   = i - row * GROUPS;           // 0..4 : adjacent lanes ->
            const float* gp = u_seq               // contiguous 16B groups
                + (size_t)(r0 + row) * HSTEPS + c * TCHUNK + g * 4;
            unsigned lofs = lds_base
                + (unsigned)((c & 1) * (BUFFLOATS * 4) + row * (STRIDEF * 4) + g * 16);
            asm volatile("global_load_async_to_lds_b128 %0, %1, off"
                         :: "v"(lofs), "v"(gp)
                         : "memory");
        }
    };

    float x1 = x1_0[b];
    float v1 = v1_0[b];
    float x2 = x2_0[b];
    float v2 = v2_0[b];

    issue(0);

    for (int c = 0; c < NCHUNK; ++c) {
        if (c + 1 < NCHUNK) {
            issue(c + 1);
            // Async loads complete in order: <=NLOADS outstanding means all
            // of chunk c's loads have landed in LDS (c+1's may be in flight).
            asm volatile("s_wait_asynccnt 5" ::: "memory");
        } else {
            asm volatile("s_wait_asynccnt 0" ::: "memory");
        }
        __syncthreads();   // everyone's chunk-c data visible

        const int base = (c & 1) * BUFFLOATS + tid * STRIDEF;
#pragma unroll
        for (int t = 0; t < TCHUNK; t += 2) {
            float2 uu = *(const float2*)(&tile[base + t]);   // ds_load_b64
            sim_step(uu.x, x1, v1, x2, v2);
            sim_step(uu.y, x1, v1, x2, v2);
        }
        __syncthreads();   // done reading buffer before chunk c+2 overwrites it
    }

    float d = x2 - goal[b];
    out[b]         = d * d;   // loss
    out[B + b]     = x1;
    out[2 * B + b] = x2;
}

extern "C" void kernel_launch(void* const* d_in, const int* in_sizes, int n_in,
                              void* d_out, int out_size, void* d_ws, size_t ws_size,
                              hipStream_t stream) {
    const float* u_seq = (const float*)d_in[0];
    const float* x1_0  = (const float*)d_in[1];
    const float* v1_0  = (const float*)d_in[2];
    const float* x2_0  = (const float*)d_in[3];
    const float* v2_0  = (const float*)d_in[4];
    const float* goal  = (const float*)d_in[5];
    float* out = (float*)d_out;

    const int B = in_sizes[1];          // 131072, divisible by 256
    const int grid = B / ROWS;

    pushing_env_scan_kernel<<<grid, ROWS, 0, stream>>>(
        u_seq, x1_0, v1_0, x2_0, v2_0, goal, out, B);
}